// SDFGrid_6682969113121
// MI455X (gfx1250) — compile-verified
//
#include <hip/hip_runtime.h>

// ---------------------------------------------------------------------------
// SDF grid: fused normal-grid + trilinear-interp + relu(-g) gather kernel.
//
// Roofline: no matmul structure -> WMMA/TDM inapplicable. 1.28M points,
// ~56B/point streaming + 64MB grid that stays resident in MI455X's 192MB L2.
// HBM floor ~(72MB + 64MB)/23.3TB/s ~ 6us. The win is (a) fusing instead of
// materializing 3x64MB normal grids like the reference (+400MB traffic), and
// (b) issuing the minimal 32 unique L2 gathers per point up front.
// ---------------------------------------------------------------------------

#define NTHREADS 256

// VOXEL_SIZE = 4/255 ; 1/(2*vs) = 255/8 = 31.875 (exactly representable)
#define VSIZE   (4.0f / 255.0f)
#define DSCALE  31.875f

__global__ void __launch_bounds__(NTHREADS)
sdf_grid_kernel(const float* __restrict__ grid,   // [256,256,256]
                const int*   __restrict__ vidx,   // [N,3]
                const float* __restrict__ ipos,   // [N,3]
                const float* __restrict__ vmin,   // [N,3]
                const int*   __restrict__ mask,   // [N]
                float4*      __restrict__ out,    // [N] -> (nx,ny,nz,gx)
                int npts) {
  const int pid = blockIdx.x * blockDim.x + threadIdx.x;
  if (pid >= npts) return;

  // gfx1250 global_prefetch_b8 (TH=0 = speculative: OOB addresses are
  // silently dropped, so no guard branch needed).
  const int pf = pid + 16 * NTHREADS;
  __builtin_prefetch(&vidx[3 * pf], 0, 0);
  __builtin_prefetch(&ipos[3 * pf], 0, 0);
  __builtin_prefetch(&vmin[3 * pf], 0, 0);
  __builtin_prefetch(&mask[pf],     0, 0);

  const int x = vidx[3 * pid + 0];
  const int y = vidx[3 * pid + 1];
  const int z = vidx[3 * pid + 2];

  const float tx = (ipos[3 * pid + 0] - vmin[3 * pid + 0]) / VSIZE;
  const float ty = (ipos[3 * pid + 1] - vmin[3 * pid + 1]) / VSIZE;
  const float tz = (ipos[3 * pid + 2] - vmin[3 * pid + 2]) / VSIZE;

  const float m   = (float)mask[pid];
  const float inv = 1.0f - m;          // +1 on every corner when masked out

  // ---- minimal unique gather set: 8 core + 6 x 4 extension values --------
  const int sx = 1 << 16, sy = 1 << 8;
  const int b  = z + (y << 8) + (x << 16);

  const bool bxl = (x > 0), bxh = (x < 254);
  const bool byl = (y > 0), byh = (y < 254);
  const bool bzl = (z > 0), bzh = (z < 254);

  // extension offsets (elements); at the edge both selectors fall back to
  // the inner value used by the one-sided (3*edge - inner)/2 formula.
  const int exl = bxl ? -sx : 2 * sx;
  const int exh = bxh ? 2 * sx : -sx;
  const int eyl = byl ? -sy : 2 * sy;
  const int eyh = byh ? 2 * sy : -sy;
  const int ezl = bzl ? -1 : 2;
  const int ezh = bzh ? 2 : -1;

  float core[2][2][2], xl[2][2], xh[2][2], yl[2][2], yh[2][2], zl[2][2], zh[2][2];

#pragma unroll
  for (int dy = 0; dy < 2; ++dy)
#pragma unroll
    for (int dz = 0; dz < 2; ++dz) {
      const int o = b + dz + (dy << 8);
      core[0][dy][dz] = grid[o];
      core[1][dy][dz] = grid[o + sx];
      xl[dy][dz]      = grid[o + exl];
      xh[dy][dz]      = grid[o + exh];
    }
#pragma unroll
  for (int dx = 0; dx < 2; ++dx)
#pragma unroll
    for (int dz = 0; dz < 2; ++dz) {
      const int o = b + dz + (dx << 16);
      yl[dx][dz] = grid[o + eyl];
      yh[dx][dz] = grid[o + eyh];
    }
#pragma unroll
  for (int dx = 0; dx < 2; ++dx)
#pragma unroll
    for (int dy = 0; dy < 2; ++dy) {
      const int o = b + (dy << 8) + (dx << 16);
      zl[dx][dy] = grid[o + ezl];
      zh[dx][dy] = grid[o + ezh];
    }

  // ---- per-corner derivatives (pure VALU on registers) + trilinear -------
  float n0 = 0.0f, n1 = 0.0f, n2 = 0.0f;

#pragma unroll
  for (int dx = 0; dx < 2; ++dx)
#pragma unroll
    for (int dy = 0; dy < 2; ++dy)
#pragma unroll
      for (int dz = 0; dz < 2; ++dz) {
        const float wx = dx ? tx : 1.0f - tx;
        const float wy = dy ? ty : 1.0f - ty;
        const float wz = dz ? tz : 1.0f - tz;
        const float w  = wz * wy * wx;   // association matches reference

        float fwd, bwd;

        // x-derivative at corner (x+dx, y+dy, z+dz)
        if (dx == 0) {
          fwd = core[1][dy][dz];
          bwd = bxl ? xl[dy][dz]
                    : 0.5f * (3.0f * core[0][dy][dz] - xl[dy][dz]);
        } else {
          fwd = bxh ? xh[dy][dz]
                    : 0.5f * (3.0f * core[1][dy][dz] - xh[dy][dz]);
          bwd = core[0][dy][dz];
        }
        n0 += w * ((fwd - bwd) * DSCALE + inv);

        // y-derivative
        if (dy == 0) {
          fwd = core[dx][1][dz];
          bwd = byl ? yl[dx][dz]
                    : 0.5f * (3.0f * core[dx][0][dz] - yl[dx][dz]);
        } else {
          fwd = byh ? yh[dx][dz]
                    : 0.5f * (3.0f * core[dx][1][dz] - yh[dx][dz]);
          bwd = core[dx][0][dz];
        }
        n1 += w * ((fwd - bwd) * DSCALE + inv);

        // z-derivative
        if (dz == 0) {
          fwd = core[dx][dy][1];
          bwd = bzl ? zl[dx][dy]
                    : 0.5f * (3.0f * core[dx][dy][0] - zl[dx][dy]);
        } else {
          fwd = bzh ? zh[dx][dy]
                    : 0.5f * (3.0f * core[dx][dy][1] - zh[dx][dy]);
          bwd = core[dx][dy][0];
        }
        n2 += w * ((fwd - bwd) * DSCALE + inv);
      }

  // relu(-grid) at the base corner (== core[0][0][0], no extra load),
  // zeroed where mask==0
  const float gx = fmaxf(-core[0][0][0], 0.0f) * m;

  out[pid] = make_float4(n0, n1, n2, gx);
}

extern "C" void kernel_launch(void* const* d_in, const int* in_sizes, int n_in,
                              void* d_out, int out_size, void* d_ws, size_t ws_size,
                              hipStream_t stream) {
  (void)n_in; (void)d_ws; (void)ws_size; (void)out_size;

  const float* grid = (const float*)d_in[0];   // [256^3] f32
  const int*   vidx = (const int*)  d_in[1];   // [N,3]   i32
  const float* ipos = (const float*)d_in[2];   // [N,3]   f32
  const float* vmin = (const float*)d_in[3];   // [N,3]   f32
  const int*   mask = (const int*)  d_in[4];   // [N]     i32

  const int npts = in_sizes[4];                // H*W (mask element count)
  const int blocks = (npts + NTHREADS - 1) / NTHREADS;

  sdf_grid_kernel<<<blocks, NTHREADS, 0, stream>>>(
      grid, vidx, ipos, vmin, mask, (float4*)d_out, npts);
}